// TRM_57286273794231
// MI455X (gfx1250) — compile-verified
//
#include <hip/hip_runtime.h>
#include <hip/hip_bf16.h>
#include <math.h>

// ---------------------------------------------------------------------------
// TRM recurrent net on MI455X (gfx1250, wave32).
//
// Per wave: 16 batch elements in the N dimension of V_WMMA_F32_16X16X4_F32
// tiles. A = transposed 4x4 blocks of Wnet (M = output feature), B = state
// (K = input feature, N = batch), C = cached affine terms / bias.
//
// Row-duplication trick (avoids all cross-lane traffic):
//   A rows 8..11 duplicate rows 0..3 (rows 4..7, 12..15 zero), and C rows
//   8..11 duplicate the bias rows. Then D row 8+j == D row j, i.e. D VGPR j
//   holds feature j of batch column (lane&15) in EVERY lane. The D -> B
//   repack is then two v_cndmask picks (B.x = lo ? d0 : d2, B.y = lo ? d1:d3)
//   done BEFORE tanh, so each inner step is:
//       1 x v_wmma_f32_16x16x4_f32  +  2 x v_cndmask  +  2 x v_tanh_f32
//
// Layouts assumed (CDNA5 ISA 7.12.2):
//   A (16x4 f32):  lane l -> M = l&15 ; VGPR0 = K = 2*(l>>4), VGPR1 = K+1
//   B (4x16 f32):  lane l -> N = l&15 ; VGPR0 = K = 2*(l>>4), VGPR1 = K+1
//   C/D (16x16 f32): VGPR g, lanes 0-15 -> (M=g, N=lane); lanes 16-31 -> M=g+8
// ---------------------------------------------------------------------------

typedef __attribute__((ext_vector_type(2))) float v2f;
typedef __attribute__((ext_vector_type(8))) float v8f;

__device__ __forceinline__ float trm_tanh(float x) {
#if __has_builtin(__builtin_amdgcn_tanhf)
  return __builtin_amdgcn_tanhf(x);   // v_tanh_f32 (gfx1250 TRANS op)
#else
  return tanhf(x);
#endif
}

__device__ __forceinline__ v8f wmma4(v2f a, v2f b, v8f c) {
  // (neg_a, A, neg_b, B, c_mod, C, reuse_a, reuse_b)
  return __builtin_amdgcn_wmma_f32_16x16x4_f32(false, a, false, b, (short)0, c,
                                               false, false);
}

// Core recurrence. TT/NN == 0 means "use dynamic rt_T / rt_n".
template <int TT, int NN>
__device__ __forceinline__ void trm_core(int rt_T, int rt_n, bool lo,
                                         v2f A2, v2f A3, v8f cb, v8f cx,
                                         v2f Bye, v8f cxy,
                                         float& t0, float& t1, float& t2,
                                         float& t3) {
  const int Touter = TT ? TT : rt_T;
  const int Ninner = NN ? NN : rt_n;
  v2f Bz;
  Bz.x = 0.0f;
  Bz.y = 0.0f;
#pragma unroll 1
  for (int t = 0; t < Touter; ++t) {
    // ---- n inner z-steps: z = tanh(W3^T z + cxy)
#pragma unroll
    for (int i = 0; i < (NN ? NN : 1) * (NN ? 1 : 1); ++i) {
      if (!NN && i) break;  // (placeholder; real dynamic loop below)
      (void)i;
      break;
    }
    if (NN) {
#pragma unroll
      for (int i = 0; i < NN; ++i) {
        v8f d = wmma4(A3, Bz, cxy);
        float px = lo ? d[0] : d[2];
        float py = lo ? d[1] : d[3];
        Bz.x = trm_tanh(px);
        Bz.y = trm_tanh(py);
      }
    } else {
#pragma unroll 1
      for (int i = 0; i < Ninner; ++i) {
        v8f d = wmma4(A3, Bz, cxy);
        float px = lo ? d[0] : d[2];
        float py = lo ? d[1] : d[3];
        Bz.x = trm_tanh(px);
        Bz.y = trm_tanh(py);
      }
    }
    // ---- ye = tanh(W2^T ye + W3^T z + bnet)   (x contribution is zero)
    v8f u  = wmma4(A3, Bz, cb);
    v8f u2 = wmma4(A2, Bye, u);
    t0 = trm_tanh(u2[0]);
    t1 = trm_tanh(u2[1]);
    t2 = trm_tanh(u2[2]);   // valid in all lanes thanks to row duplication
    t3 = trm_tanh(u2[3]);
    Bye.x = lo ? t0 : t2;
    Bye.y = lo ? t1 : t3;
    // refresh cached affine term with the new ye
    cxy = wmma4(A2, Bye, cx);
  }
}

__global__ __launch_bounds__(256) void trm_wmma_kernel(
    const float* __restrict__ xs,
    const float* __restrict__ Wx,   const float* __restrict__ bx,
    const float* __restrict__ by,
    const float* __restrict__ Wnet, const float* __restrict__ bnet,
    const float* __restrict__ Wdec, const float* __restrict__ bdec,
    const float* __restrict__ Wq,   const float* __restrict__ bq,
    const int* __restrict__ Tp,     const int* __restrict__ np,
    float* __restrict__ out, int Bn) {
  const int lane = threadIdx.x & 31;
  const int wave = blockIdx.x * (blockDim.x >> 5) + (threadIdx.x >> 5);
  const int base = wave * 16;
  if (base >= Bn) return;  // wave-uniform exit: EXEC stays all-ones below

  const int  n    = lane & 15;   // batch column / A-C row M = lane&15
  const int  kh   = lane >> 4;   // K half (0: K=0,1 ; 1: K=2,3)
  const bool lo   = (kh == 0);
  const int  elem = min(base + n, Bn - 1);

  // ---- x encoder (scalar per lane; both lane halves hold the same element)
  const float x0 = xs[2 * elem + 0];
  const float x1 = xs[2 * elem + 1];
  float xe[4];
#pragma unroll
  for (int j = 0; j < 4; ++j)
    xe[j] = fmaf(x1, Wx[4 + j], fmaf(x0, Wx[j], bx[j]));

  // ---- A operands: transposed 4x4 blocks of Wnet.
  //      Rows 0-3 and 8-11 hold the weights (duplicated); 4-7, 12-15 zero.
  const int  k0 = 2 * kh, k1 = 2 * kh + 1;
  const bool mv = ((n & 7) < 4);   // rows 0-3 and 8-11
  const int  r  = n & 3;           // duplicated weight row index
  v2f A1, A2, A3;
  A1.x = mv ? Wnet[(0 + k0) * 4 + r] : 0.0f;
  A1.y = mv ? Wnet[(0 + k1) * 4 + r] : 0.0f;
  A2.x = mv ? Wnet[(4 + k0) * 4 + r] : 0.0f;
  A2.y = mv ? Wnet[(4 + k1) * 4 + r] : 0.0f;
  A3.x = mv ? Wnet[(8 + k0) * 4 + r] : 0.0f;
  A3.y = mv ? Wnet[(8 + k1) * 4 + r] : 0.0f;

  // ---- C bias tile: rows 0-3 AND 8-11 = bnet (duplicated), rest zero.
  //      C VGPR g covers rows g (lanes 0-15) and g+8 (lanes 16-31), so the
  //      duplication means VGPRs 0-3 = bnet[g] in ALL lanes, VGPRs 4-7 = 0.
  v8f cb;
#pragma unroll
  for (int g = 0; g < 8; ++g) cb[g] = (g < 4) ? bnet[g] : 0.0f;

  // ---- B operands: xe (constant), ye (init = by, since ys = 0)
  v2f Bxe; Bxe.x = lo ? xe[0] : xe[2]; Bxe.y = lo ? xe[1] : xe[3];
  const float by0 = by[0], by1 = by[1], by2 = by[2], by3 = by[3];
  v2f Bye; Bye.x = lo ? by0 : by2; Bye.y = lo ? by1 : by3;

  // cx  = W1^T xe + bnet  (fixed);  cxy = W2^T ye + cx (refreshed per outer)
  v8f cx  = wmma4(A1, Bxe, cb);
  v8f cxy = wmma4(A2, Bye, cx);

  const int T  = *Tp;
  const int NN = *np;

  float t0 = by0, t1 = by1, t2 = by2, t3 = by3;  // ye for T==0 edge case

  if (T == 16 && NN == 6) {
    // hot path: compile-time trip counts -> clean unrolled WMMA chain
    trm_core<16, 6>(T, NN, lo, A2, A3, cb, cx, Bye, cxy, t0, t1, t2, t3);
  } else if (T > 0) {
    trm_core<0, 0>(T, NN, lo, A2, A3, cb, cx, Bye, cxy, t0, t1, t2, t3);
  }

  // ---- heads: lanes 0-15 hold the complete final ye (t0..t3) per element
  float y = bdec[0];
  y = fmaf(t0, Wdec[0], y); y = fmaf(t1, Wdec[1], y);
  y = fmaf(t2, Wdec[2], y); y = fmaf(t3, Wdec[3], y);
  float qa = bq[0];
  qa = fmaf(t0, Wq[0], qa); qa = fmaf(t1, Wq[1], qa);
  qa = fmaf(t2, Wq[2], qa); qa = fmaf(t3, Wq[3], qa);
  float q = 1.0f / (1.0f + __expf(-qa));

  if (lo && (base + n) < Bn) {
    out[base + n]      = y;  // y_hat block
    out[Bn + base + n] = q;  // q_hat block
  }
}

extern "C" void kernel_launch(void* const* d_in, const int* in_sizes, int n_in,
                              void* d_out, int out_size, void* d_ws,
                              size_t ws_size, hipStream_t stream) {
  // setup_inputs order:
  // 0:xs 1:Wx 2:bx 3:Wy 4:by 5:Wnet 6:bnet 7:Wdec 8:bdec 9:Wq 10:bq 11:T 12:n
  const float* xs   = (const float*)d_in[0];
  const float* Wx   = (const float*)d_in[1];
  const float* bx   = (const float*)d_in[2];
  const float* by   = (const float*)d_in[4];   // Wy unused: ys is all-zero
  const float* Wnet = (const float*)d_in[5];
  const float* bnet = (const float*)d_in[6];
  const float* Wdec = (const float*)d_in[7];
  const float* bdec = (const float*)d_in[8];
  const float* Wq   = (const float*)d_in[9];
  const float* bq   = (const float*)d_in[10];
  const int*   Tp   = (const int*)d_in[11];
  const int*   np   = (const int*)d_in[12];
  float*       out  = (float*)d_out;

  const int Bn      = in_sizes[0] / 2;            // xs is [B, 2]
  const int waves   = (Bn + 15) / 16;             // 16 elements per wave32
  const int threads = 256;                        // 8 waves per block
  const int blocks  = (waves * 32 + threads - 1) / threads;

  trm_wmma_kernel<<<blocks, threads, 0, stream>>>(
      xs, Wx, bx, by, Wnet, bnet, Wdec, bdec, Wq, bq, Tp, np, out, Bn);
}